// Encoder_76046690943032
// MI455X (gfx1250) — compile-verified
//
#include <hip/hip_runtime.h>

// ---------------------------------------------------------------------------
// Two-phase LSTM encoder for MI455X (gfx1250, wave32).
//   B=524288 independent elements, E=16, H=8, T_OBS=8, T_PRE=12.
// Strategy:
//   * Fold input embedding into LSTM input weights: Wx = W_ih @ W_in (32x2),
//     bfold = W_ih @ b_in + b_ih + b_hh.  gates = [h(8) | x0 x1 1 0] @ B(12x32).
//   * Sigmoid prescale (x -> 0.5x) baked into the i/f/o gate columns so every
//     activation is one v_tanh_f32 (+1 fma for sigmoids).
//   * One wave owns a 16-element batch tile. Per step: 6x v_wmma_f32_16x16x4_f32
//     (2 gate tiles x 3 K-chunks), ONE ds_swizzle per D-register to combine
//     i/g-lane and f/o-lane products, LDS transpose of new h back into WMMA
//     A-layout for the recurrence.
//   * c/h state lives in the f/o lanes (lane&8 != 0).
// ---------------------------------------------------------------------------

typedef float v2f __attribute__((ext_vector_type(2)));
typedef float v8f __attribute__((ext_vector_type(8)));

#define T_OBS 8
#define T_PRE 12
#define BATCH 524288
#define HDIM  8

__device__ __forceinline__ float fast_tanh(float x) {
#if __has_builtin(__builtin_amdgcn_tanhf)
  return __builtin_amdgcn_tanhf(x);          // v_tanh_f32 (gfx1250 trans op)
#else
  float e = __expf(2.0f * x);
  return 1.0f - 2.0f * __builtin_amdgcn_rcpf(e + 1.0f);
#endif
}

// swap lanes d <-> d+8 within each group of 16 (ds_swizzle xor=8, and=0x1f)
__device__ __forceinline__ float swap8(float x) {
  return __int_as_float(__builtin_amdgcn_ds_swizzle(__float_as_int(x), 0x201F));
}

__device__ __forceinline__ v8f wmma4(v2f a, v2f b, v8f c) {
  // D = A(16x4) * B(4x16) + C, full fp32.
  return __builtin_amdgcn_wmma_f32_16x16x4_f32(false, a, false, b, (short)0, c,
                                               false, false);
}

// Build per-lane B-matrix fragments for one phase.
// Bw[tile][chunk]: tile0 = gate cols 0..15 (i,f), tile1 = 16..31 (g,o).
// chunk0/1 = W_hh^T rows 0..3 / 4..7 ; chunk2 rows = [Wx_col0, Wx_col1, bfold, 0].
// B 4x16 layout: V0 = rows {k0 (lanes0-15), k2 (lanes16-31)}, V1 = {k1, k3}.
// Sigmoid gates (i,f,o) get their whole column scaled by 0.5.
__device__ __forceinline__ void load_weights(
    const float* __restrict__ W_in, const float* __restrict__ b_in,
    const float* __restrict__ W_ih, const float* __restrict__ W_hh,
    const float* __restrict__ b_ih, const float* __restrict__ b_hh,
    v2f Bw[2][3], int lane) {
  const int half = lane >> 4;
#pragma unroll
  for (int tile = 0; tile < 2; ++tile) {
    const int row = tile * 16 + (lane & 15);   // gate column index (0..31)
    // gates 16..23 are 'g' (tanh, unscaled); all others sigmoid -> 0.5x
    const float s = (row >= 16 && row < 24) ? 1.0f : 0.5f;
#pragma unroll
    for (int ck = 0; ck < 2; ++ck) {           // h-dim chunks
      Bw[tile][ck].x = s * W_hh[row * 8 + ck * 4 + half * 2 + 0];
      Bw[tile][ck].y = s * W_hh[row * 8 + ck * 4 + half * 2 + 1];
    }
    float wx0 = 0.0f, wx1 = 0.0f, bf = b_ih[row] + b_hh[row];
#pragma unroll
    for (int e = 0; e < 16; ++e) {             // fold embedding layer
      const float w = W_ih[row * 16 + e];
      wx0 += w * W_in[e * 2 + 0];
      wx1 += w * W_in[e * 2 + 1];
      bf  += w * b_in[e];
    }
    Bw[tile][2].x = half ? (s * bf) : (s * wx0);   // rows [x0 | 1]
    Bw[tile][2].y = half ? 0.0f     : (s * wx1);   // rows [x1 | 0]
  }
}

// c-state layout: c[r] valid at lanes 8-15 (batch r, dim=lane&7) and 24-31
// (batch r+8) -- the f/o lanes of the D tiles.
__device__ __forceinline__ void load_c(const float* __restrict__ cptr,
                                       float c[8], int lane, int tileBase) {
  const int d = lane & 7, half = lane >> 4;
  const bool valid = (lane & 8) != 0;
#pragma unroll
  for (int r = 0; r < 8; ++r) {
    float v = 0.0f;
    if (valid) v = cptr[(size_t)(tileBase + half * 8 + r) * HDIM + d];
    c[r] = v;
  }
}

__device__ __forceinline__ void store_h(float* __restrict__ out,
                                        const float hreg[8], int lane,
                                        int tileBase) {
  const int d = lane & 7, half = lane >> 4;
  if ((lane & 8) != 0) {
    const size_t base = (size_t)d * BATCH + tileBase + half * 8;
#pragma unroll
    for (int r = 0; r < 8; ++r) out[base + r] = hreg[r];   // out[d*B + b]
  }
}

template <int T>
__device__ __forceinline__ void lstm_phase(
    const float* __restrict__ traj,     // [T, BATCH, 2]
    const v2f Bw[2][3],
    v2f& A0, v2f& A1,                   // h state in WMMA A layout (in/out)
    float c[8], float hreg[8],
    float* __restrict__ ldsw,           // per-wave 256-float LDS region
    int lane, int tileBase) {
  const int m = lane & 15;
  const int half = lane >> 4;
  const int d = lane & 7;
  const bool valid = (lane & 8) != 0;   // f/o lanes hold state
  // LDS row stride 10 dwords keeps b64 loads 8B-aligned; invalid lanes dump
  // into trash rows 160..245 inside this wave's private 256-dword region.
  const int storeBase = valid ? (half * 80 + d) : (160 + half * 8 + d);
  const int aload = m * 10 + half * 2;

  for (int t = 0; t < T; ++t) {
    // A chunk2: [x0 x1] for lanes 0-15, [1 0] for lanes 16-31 (bias row)
    const v2f x = *(const v2f*)(traj + ((size_t)t * BATCH + tileBase + m) * 2);
    v2f A2;
    A2.x = half ? 1.0f : x.x;
    A2.y = half ? 0.0f : x.y;

    const v8f z = {0.f, 0.f, 0.f, 0.f, 0.f, 0.f, 0.f, 0.f};
    v8f D0 = wmma4(A0, Bw[0][0], wmma4(A1, Bw[0][1], wmma4(A2, Bw[0][2], z)));
    v8f D1 = wmma4(A0, Bw[1][0], wmma4(A1, Bw[1][1], wmma4(A2, Bw[1][2], z)));

#pragma unroll
    for (int r = 0; r < 8; ++r) {
      // D layout: VGPR r -> batch {r | r+8}, lane -> gate column.
      // t0: 0.5*i (lanes d), 0.5*f (lanes d+8).  t1: g (lanes d), 0.5*o (d+8).
      const float tv0 = fast_tanh(D0[r]);
      const float sif = tv0 * 0.5f + 0.5f;   // sigmoid(i) low / sigmoid(f) high
      const float tv1 = fast_tanh(D1[r]);    // tanh(g) low / tanh(0.5*o) high
      const float oo  = tv1 * 0.5f + 0.5f;   // sigmoid(o) valid in high lanes
      const float P   = sif * tv1;           // sig(i)*tanh(g) valid low lanes
      const float Psw = swap8(P);            // bring into f/o lanes
      const float cn  = sif * c[r] + Psw;    // sig(f)*c + sig(i)*tanh(g)
      c[r] = cn;
      const float hh = oo * fast_tanh(cn);   // sig(o)*tanh(c)
      hreg[r] = hh;
      ldsw[storeBase + r * 10] = hh;         // transpose staging
    }
    __builtin_amdgcn_wave_barrier();
    // Reload h in A layout: lane m -> dims {0,1}; lane m+16 -> dims {2,3};
    // +4 dwords -> dims {4,5}/{6,7}
    A0 = *(const v2f*)(ldsw + aload);
    A1 = *(const v2f*)(ldsw + aload + 4);
    __builtin_amdgcn_wave_barrier();
  }
}

__global__ __launch_bounds__(256) void lstm_encoder_kernel(
    const float* __restrict__ obs, const float* __restrict__ pre,
    const float* __restrict__ h0, const float* __restrict__ c0,
    const float* __restrict__ c0p, const float* __restrict__ W_in,
    const float* __restrict__ b_in, const float* __restrict__ Wiho,
    const float* __restrict__ Whho, const float* __restrict__ biho,
    const float* __restrict__ bhho, const float* __restrict__ Wihp,
    const float* __restrict__ Whhp, const float* __restrict__ bihp,
    const float* __restrict__ bhhp, float* __restrict__ out) {
  __shared__ float lds[8 * 256];
  const int lane = threadIdx.x & 31;
  const int wave = threadIdx.x >> 5;
  float* ldsw = lds + wave * 256;
  const int tileBase = (blockIdx.x * 8 + wave) * 16;

  v2f Bo[2][3], Bp[2][3];
  load_weights(W_in, b_in, Wiho, Whho, biho, bhho, Bo, lane);
  load_weights(W_in, b_in, Wihp, Whhp, bihp, bhhp, Bp, lane);

  // h0 directly in A layout: lane m -> dims {0,1}, lane m+16 -> {2,3}; +4 -> {4..7}
  v2f A0, A1;
  {
    const float* hb =
        h0 + (size_t)(tileBase + (lane & 15)) * HDIM + (lane >> 4) * 2;
    A0 = *(const v2f*)hb;
    A1 = *(const v2f*)(hb + 4);
  }

  float c[8], hreg[8];
  load_c(c0, c, lane, tileBase);
  lstm_phase<T_OBS>(obs, Bo, A0, A1, c, hreg, ldsw, lane, tileBase);
  store_h(out, hreg, lane, tileBase);                 // c_out = h^T flat

  load_c(c0p, c, lane, tileBase);                     // cell re-initialized
  lstm_phase<T_PRE>(pre, Bp, A0, A1, c, hreg, ldsw, lane, tileBase);
  store_h(out + (size_t)BATCH * HDIM, hreg, lane, tileBase);  // x_out
}

extern "C" void kernel_launch(void* const* d_in, const int* in_sizes, int n_in,
                              void* d_out, int out_size, void* d_ws,
                              size_t ws_size, hipStream_t stream) {
  (void)in_sizes; (void)n_in; (void)out_size; (void)d_ws; (void)ws_size;
  const int tiles = BATCH / 16;        // 32768 waves
  const int blocks = tiles / 8;        // 8 waves (256 threads) per block
  lstm_encoder_kernel<<<blocks, 256, 0, stream>>>(
      (const float*)d_in[0], (const float*)d_in[1], (const float*)d_in[2],
      (const float*)d_in[3], (const float*)d_in[4], (const float*)d_in[5],
      (const float*)d_in[6], (const float*)d_in[7], (const float*)d_in[8],
      (const float*)d_in[9], (const float*)d_in[10], (const float*)d_in[11],
      (const float*)d_in[12], (const float*)d_in[13], (const float*)d_in[14],
      (float*)d_out);
}